// Diff_Pool__47296179863967
// MI455X (gfx1250) — compile-verified
//
#include <hip/hip_runtime.h>
#include <hip/hip_bf16.h>

// ---------------- CDNA5 WMMA helpers ----------------
typedef __attribute__((ext_vector_type(2))) float v2f;
typedef __attribute__((ext_vector_type(8))) float v8f;

__device__ __forceinline__ v8f wmma_k4(v2f a, v2f b, v8f c) {
  // D = A(16x4 f32) * B(4x16 f32) + C(16x16 f32)
  return __builtin_amdgcn_wmma_f32_16x16x4_f32(
      /*neg_a=*/false, a, /*neg_b=*/false, b,
      /*c_mod=*/(short)0, c, /*reuse_a=*/false, /*reuse_b=*/false);
}

__device__ __forceinline__ void atomAddF(float* p, float v) {
  unsafeAtomicAdd(p, v);  // global_atomic_add_f32 (no return)
}

// ---------------- problem constants ----------------
#define BATCHES   64
#define NPB       2048
#define BN_TOT    (BATCHES * NPB)   // 131072 nodes
#define BE_TOT    (BN_TOT * 16)     // 2097152 edges
#define MCL       16                // clusters
#define C1        64
#define C2        64
#define KEDGE     4
#define WAVES     8                 // waves per block (wave32)
#define TILES     (BN_TOT / 16)     // 8192 node tiles
#define TILES_PER_BATCH (NPB / 16)  // 128
#define BLOCKS_PER_BATCH (TILES_PER_BATCH / WAVES)  // 16 -> block is batch-uniform

// =====================================================================
// Kernel 1: per-16-node tile (one wave each):
//   logits = X@Wa ; S = softmax(logits) -> S_buf + LDS
//   G = [p | X@We + be] (16x64) -> LDS
//   Z_blk += S^T @ G and w_blk += colsum(S), reduced across the 8 waves
//   in LDS (ds_add_f32), then one block-wide global f32 atomic pass.
// =====================================================================
__global__ __launch_bounds__(256) void k_node(
    const float* __restrict__ x,
    const float* __restrict__ Wa, const float* __restrict__ ba,
    const float* __restrict__ We, const float* __restrict__ be,
    float* __restrict__ S_buf, float* __restrict__ Z_buf,
    float* __restrict__ w_buf)
{
  __shared__ float sS[WAVES][16][MCL];   // [node-in-tile][cluster]   8 KB
  __shared__ float sG[WAVES][16][C2];    // [node-in-tile][out col]  32 KB
  __shared__ float zAcc[MCL][C2];        // block Z accumulator       4 KB
  __shared__ float wAcc[MCL];

  const int wave = threadIdx.x >> 5;
  const int lane = threadIdx.x & 31;
  const int n    = lane & 15;        // N index / A-row index
  const int half = lane >> 4;        // 0: K=0,1  1: K=2,3 (per K-chunk)
  const int koff = half * 2;
  const int tile = blockIdx.x * WAVES + wave;
  const int base = tile * 16;
  const int batch = blockIdx.x / BLOCKS_PER_BATCH;  // uniform per block

  // zero block accumulators
  for (int i = threadIdx.x; i < MCL * C2; i += 256) ((float*)zAcc)[i] = 0.f;
  if (threadIdx.x < MCL) wAcc[threadIdx.x] = 0.f;

  // ---- A-matrix: X tile in 16x4-f32 layout, hoisted for all 5 GEMMs ----
  const float* xrow = x + (size_t)(base + n) * C1;   // lane%16 == A row m
  v2f areg[16];
  #pragma unroll
  for (int k = 0; k < 16; ++k)
    areg[k] = ((const v2f*)xrow)[2 * k + half];      // {X[m][4k+koff], X[m][4k+koff+1]}

  // ---- logits = X @ W_assign (64x16): 16 WMMAs ----
  v8f acc = {};
  #pragma unroll
  for (int k = 0; k < 16; ++k) {
    const int k0 = 4 * k + koff;
    v2f b; b.x = Wa[k0 * MCL + n]; b.y = Wa[(k0 + 1) * MCL + n];
    acc = wmma_k4(areg[k], b, acc);
  }

  // ---- softmax over clusters (N dim == 16 lanes of our half) ----
  const float bias = ba[n];
  float sv[8];
  #pragma unroll
  for (int v = 0; v < 8; ++v) {
    float l = acc[v] + bias;
    float mx = l;
    #pragma unroll
    for (int msk = 1; msk < 16; msk <<= 1)
      mx = fmaxf(mx, __shfl_xor(mx, msk, 32));
    float e = __expf(l - mx);
    float s = e;
    #pragma unroll
    for (int msk = 1; msk < 16; msk <<= 1)
      s += __shfl_xor(s, msk, 32);
    sv[v] = e / s;
  }

  // ---- publish S (global for edge/A_MM kernels, LDS for S^T GEMM) ----
  #pragma unroll
  for (int v = 0; v < 8; ++v) {
    const int row = v + 8 * half;               // node within tile
    S_buf[(size_t)(base + row) * MCL + n] = sv[v];
    sS[wave][row][n] = sv[v];
  }

  // ---- G = [p(3) | f(61)] : 4 column tiles of 16, 16 WMMAs each ----
  #pragma unroll
  for (int j = 0; j < 4; ++j) {
    const int nc = j * 16 + n;                  // output column 0..63
    v8f g = {};
    #pragma unroll
    for (int k = 0; k < 16; ++k) {
      const int k0 = 4 * k + koff;
      v2f b;
      b.x = (nc >= 3) ? We[(size_t)k0       * (C2 - 3) + (nc - 3)] : 0.f;
      b.y = (nc >= 3) ? We[(size_t)(k0 + 1) * (C2 - 3) + (nc - 3)] : 0.f;
      g = wmma_k4(areg[k], b, g);
    }
    const float gb = (nc >= 3) ? be[nc - 3] : 0.f;
    #pragma unroll
    for (int v = 0; v < 8; ++v) {
      const int row = v + 8 * half;
      float val = g[v] + gb;
      if (nc < 3) val = x[(size_t)(base + row) * C1 + nc];   // p columns
      sG[wave][row][nc] = val;
    }
  }
  __syncthreads();   // sS/sG visible; zAcc/wAcc initialized

  // ---- w accumulation (ds_add_f32 into block accumulator) ----
  float csum = 0.f;
  #pragma unroll
  for (int v = 0; v < 8; ++v) csum += sv[v];
  csum += __shfl_xor(csum, 16, 32);             // combine row halves
  if (half == 0) atomicAdd(&wAcc[n], csum);

  // ---- Z_blk += S^T (16x16) @ G (16x64): 4 WMMAs per column tile ----
  #pragma unroll
  for (int j = 0; j < 4; ++j) {
    v8f t = {};
    #pragma unroll
    for (int k = 0; k < 4; ++k) {
      const int k0 = 4 * k + koff;              // node index (K dim)
      v2f a; a.x = sS[wave][k0][n];      a.y = sS[wave][k0 + 1][n];       // S^T[m][i]
      v2f b; b.x = sG[wave][k0][j*16+n]; b.y = sG[wave][k0 + 1][j*16+n];
      t = wmma_k4(a, b, t);
    }
    #pragma unroll
    for (int v = 0; v < 8; ++v) {
      const int m = v + 8 * half;               // cluster row
      atomicAdd(&zAcc[m][j * 16 + n], t[v]);    // ds_add_f32
    }
  }
  __syncthreads();

  // ---- one global atomic pass per block (8x less atomic traffic) ----
  float* Zb = Z_buf + (size_t)batch * MCL * C2;
  for (int i = threadIdx.x; i < MCL * C2; i += 256)
    atomAddF(&Zb[i], ((const float*)zAcc)[i]);
  if (threadIdx.x < MCL)
    atomAddF(&w_buf[batch * MCL + threadIdx.x], wAcc[threadIdx.x]);
}

// =====================================================================
// Kernel 2: AS[dst] += S[src]  over 2M edges (dominant memory stage)
// =====================================================================
__global__ __launch_bounds__(256) void k_edge(
    const int* __restrict__ e0, const int* __restrict__ e1,
    const float* __restrict__ S_buf, float* __restrict__ AS_buf, int ne)
{
  const int e = blockIdx.x * blockDim.x + threadIdx.x;
  if (e >= ne) return;
  const int dst = e0[e], src = e1[e];
  const float4* s4 = (const float4*)(S_buf + (size_t)src * MCL);
  float* a = AS_buf + (size_t)dst * MCL;
  #pragma unroll
  for (int q = 0; q < 4; ++q) {
    const float4 v = s4[q];
    atomAddF(&a[4 * q + 0], v.x);
    atomAddF(&a[4 * q + 1], v.y);
    atomAddF(&a[4 * q + 2], v.z);
    atomAddF(&a[4 * q + 3], v.w);
  }
}

// =====================================================================
// Kernel 3: A_MM[b] += S^T(16x16) @ AS(16x16) per node tile (4 WMMAs),
// block-reduced in LDS, then one 256-thread global atomic pass.
// =====================================================================
__global__ __launch_bounds__(256) void k_amm(
    const float* __restrict__ S_buf, const float* __restrict__ AS_buf,
    float* __restrict__ A_buf)
{
  __shared__ float aAcc[MCL][MCL];   // 1 KB block accumulator

  const int wave = threadIdx.x >> 5;
  const int lane = threadIdx.x & 31;
  const int n    = lane & 15;
  const int half = lane >> 4;
  const int koff = half * 2;
  const int tile = blockIdx.x * WAVES + wave;
  const int base = tile * 16;
  const int batch = blockIdx.x / BLOCKS_PER_BATCH;  // uniform per block

  if (threadIdx.x < MCL * MCL) ((float*)aAcc)[threadIdx.x] = 0.f;
  __syncthreads();

  v8f t = {};
  #pragma unroll
  for (int k = 0; k < 4; ++k) {
    const int k0 = 4 * k + koff;                // node index (K dim)
    v2f a; a.x = S_buf [(size_t)(base + k0) * MCL + n];
           a.y = S_buf [(size_t)(base + k0 + 1) * MCL + n];   // S^T row m=n
    v2f b; b.x = AS_buf[(size_t)(base + k0) * MCL + n];
           b.y = AS_buf[(size_t)(base + k0 + 1) * MCL + n];
    t = wmma_k4(a, b, t);
  }
  #pragma unroll
  for (int v = 0; v < 8; ++v) {
    const int m = v + 8 * half;
    atomicAdd(&aAcc[m][n], t[v]);               // ds_add_f32
  }
  __syncthreads();

  float* A = A_buf + (size_t)batch * MCL * MCL;
  if (threadIdx.x < MCL * MCL)
    atomAddF(&A[threadIdx.x], ((const float*)aAcc)[threadIdx.x]);
}

// =====================================================================
// Kernel 4: finalize:  x_out = [Zp/w | Zf];  top-4 of A_MM -> e_out, b_out
// =====================================================================
__global__ __launch_bounds__(256) void k_final(
    const float* __restrict__ Z_buf, const float* __restrict__ w_buf,
    const float* __restrict__ A_buf, float* __restrict__ out)
{
  const int NX = BATCHES * MCL * C2;            // 65536 x_out elements
  const int i = blockIdx.x * blockDim.x + threadIdx.x;
  if (i < NX) {
    const int c = i & 63, r = i >> 6;           // r = b*16 + m
    float z = Z_buf[i];
    if (c < 3) z = z / fmaxf(w_buf[r], 1e-10f);
    out[i] = z;
  } else if (i < NX + BATCHES * MCL) {
    const int r = i - NX;                       // r = b*16 + m
    const int b = r >> 4, m = r & 15;
    float vals[MCL];
    #pragma unroll
    for (int q = 0; q < MCL; ++q) vals[q] = A_buf[r * MCL + q];
    const int off = b * MCL;
    #pragma unroll
    for (int kk = 0; kk < KEDGE; ++kk) {
      int best = 0; float bv = vals[0];
      #pragma unroll
      for (int q = 1; q < MCL; ++q)
        if (vals[q] > bv) { bv = vals[q]; best = q; }   // first-max tie-break
      vals[best] = -3.4e38f;
      out[NX + r * KEDGE + kk]                     = (float)(m + off);     // src
      out[NX + BATCHES*MCL*KEDGE + r * KEDGE + kk] = (float)(best + off);  // dst
    }
    out[NX + 2 * BATCHES * MCL * KEDGE + r] = (float)b;                    // b_out
  }
}

// =====================================================================
extern "C" void kernel_launch(void* const* d_in, const int* in_sizes, int n_in,
                              void* d_out, int out_size, void* d_ws, size_t ws_size,
                              hipStream_t stream) {
  const float* x  = (const float*)d_in[0];
  const int*   e_ = (const int*)  d_in[1];
  // d_in[2] = b_ (repeat(arange(B), N)) -- implicit in tile/batch math
  const float* Wa = (const float*)d_in[3];
  const float* ba = (const float*)d_in[4];
  const float* We = (const float*)d_in[5];
  const float* be = (const float*)d_in[6];
  float* out = (float*)d_out;

  // workspace layout (floats)
  float* S_buf  = (float*)d_ws;                       // BN*16
  float* AS_buf = S_buf  + (size_t)BN_TOT * MCL;      // BN*16
  float* Z_buf  = AS_buf + (size_t)BN_TOT * MCL;      // B*16*64
  float* w_buf  = Z_buf  + BATCHES * MCL * C2;        // B*16
  float* A_buf  = w_buf  + BATCHES * MCL;             // B*16*16

  // zero all accumulator regions (AS, Z, w, A_MM); S is fully overwritten
  const size_t zbytes =
      ((size_t)BN_TOT * MCL + BATCHES * MCL * C2 + BATCHES * MCL +
       BATCHES * MCL * MCL) * sizeof(float);
  hipMemsetAsync(AS_buf, 0, zbytes, stream);

  k_node<<<TILES / WAVES, 256, 0, stream>>>(x, Wa, ba, We, be,
                                            S_buf, Z_buf, w_buf);
  k_edge<<<(BE_TOT + 255) / 256, 256, 0, stream>>>(e_, e_ + BE_TOT,
                                                   S_buf, AS_buf, BE_TOT);
  k_amm<<<TILES / WAVES, 256, 0, stream>>>(S_buf, AS_buf, A_buf);

  const int nfin = BATCHES * MCL * C2 + BATCHES * MCL;
  k_final<<<(nfin + 255) / 256, 256, 0, stream>>>(Z_buf, w_buf, A_buf, out);
}